// VLAD_66365834658084
// MI455X (gfx1250) — compile-verified
//
#include <hip/hip_runtime.h>
#include <math.h>

// ---------------- problem constants ----------------
#define BATCH 4
#define HW    784          // 28*28 pixels per batch
#define DD    512          // feature dim
#define KK    64           // clusters
#define NPIX  (BATCH*HW)   // 3136 total pixels
#define EPSV  1e-12f

typedef __attribute__((ext_vector_type(2))) float v2f;
typedef __attribute__((ext_vector_type(8))) float v8f;

// ---------------------------------------------------------------------------
// Kernel 0: cn2[k] = ||centers_k||^2 ; zero colsum (BATCH*KK) and ssq (BATCH)
// ---------------------------------------------------------------------------
__global__ __launch_bounds__(256) void vlad_prep_kernel(
    const float* __restrict__ cen, float* __restrict__ cn2,
    float* __restrict__ colsum, float* __restrict__ ssq)
{
    const int t = threadIdx.x;
    if (t < KK) {
        float s = 0.f;
        const float* c = cen + (size_t)t * DD;
        for (int d = 0; d < DD; ++d) s += c[d] * c[d];
        cn2[t] = s;
    }
    if (t < BATCH * KK) colsum[t] = 0.f;
    if (t < BATCH)      ssq[t]    = 0.f;
}

// ---------------------------------------------------------------------------
// Branch-free, software-pipelined 16xDDx64 GEMM loop for one wave.
// DRS = B-matrix "d" stride in floats (compile-time so contiguous loads merge):
//   DRS == KK : B = W_assign  (D,K) row-major, element (d,n) = Wa[d*KK + n]
//   DRS == 1  : B = centers^T, element (d,n) = cen[n*DD + d] (pairs contiguous)
// xp  = per-lane A pointer (x row + kb), loads xp[d0], xp[d0+1]
// bpt = per-lane B base pointers for the 4 N-tiles; loads bpt[d0*DRS {+DRS}]
// ---------------------------------------------------------------------------
template <int DRS>
__device__ __forceinline__ void gemm16_loop(
    const float* __restrict__ xp,
    const float* __restrict__ bp0, const float* __restrict__ bp1,
    const float* __restrict__ bp2, const float* __restrict__ bp3,
    v8f& acc0, v8f& acc1, v8f& acc2, v8f& acc3, float& xn_part)
{
    v2f a, b0, b1, b2, b3;
    // prologue: load d0 = 0
    a.x  = xp[0];          a.y  = xp[1];
    b0.x = bp0[0];         b0.y = bp0[DRS];
    b1.x = bp1[0];         b1.y = bp1[DRS];
    b2.x = bp2[0];         b2.y = bp2[DRS];
    b3.x = bp3[0];         b3.y = bp3[DRS];

    for (int d0 = 0; d0 < DD - 4; d0 += 4) {
        // prefetch next stage while current WMMAs issue
        v2f an, b0n, b1n, b2n, b3n;
        const int dn = d0 + 4;
        an.x  = xp[dn];                  an.y  = xp[dn + 1];
        b0n.x = bp0[(size_t)dn * DRS];   b0n.y = bp0[(size_t)dn * DRS + DRS];
        b1n.x = bp1[(size_t)dn * DRS];   b1n.y = bp1[(size_t)dn * DRS + DRS];
        b2n.x = bp2[(size_t)dn * DRS];   b2n.y = bp2[(size_t)dn * DRS + DRS];
        b3n.x = bp3[(size_t)dn * DRS];   b3n.y = bp3[(size_t)dn * DRS + DRS];

        xn_part += a.x * a.x + a.y * a.y;
        acc0 = __builtin_amdgcn_wmma_f32_16x16x4_f32(false, a, false, b0, (short)0, acc0, false, false);
        acc1 = __builtin_amdgcn_wmma_f32_16x16x4_f32(false, a, false, b1, (short)0, acc1, false, false);
        acc2 = __builtin_amdgcn_wmma_f32_16x16x4_f32(false, a, false, b2, (short)0, acc2, false, false);
        acc3 = __builtin_amdgcn_wmma_f32_16x16x4_f32(false, a, false, b3, (short)0, acc3, false, false);

        a = an; b0 = b0n; b1 = b1n; b2 = b2n; b3 = b3n;
    }
    // epilogue
    xn_part += a.x * a.x + a.y * a.y;
    acc0 = __builtin_amdgcn_wmma_f32_16x16x4_f32(false, a, false, b0, (short)0, acc0, false, false);
    acc1 = __builtin_amdgcn_wmma_f32_16x16x4_f32(false, a, false, b1, (short)0, acc1, false, false);
    acc2 = __builtin_amdgcn_wmma_f32_16x16x4_f32(false, a, false, b2, (short)0, acc2, false, false);
    acc3 = __builtin_amdgcn_wmma_f32_16x16x4_f32(false, a, false, b3, (short)0, acc3, false, false);
}

// ---------------------------------------------------------------------------
// Kernel 1: per 16-pixel tile compute softmax assignment and residual scale
//   wave 0: logits tile = X(16xD) * W_assign(DxK)       via WMMA f32 16x16x4
//   wave 1: dotc  tile  = X(16xD) * centers^T(DxK)      via WMMA f32 16x16x4
// then softmax over K, s = a * rsqrt(max(a^2 * ||x-c||^2, EPS)), write S,
// and atomically accumulate per-(batch,k) column sums of S.
// ---------------------------------------------------------------------------
__global__ __launch_bounds__(64) void vlad_assign_kernel(
    const float* __restrict__ x,   const float* __restrict__ Wa,
    const float* __restrict__ ba,  const float* __restrict__ cen,
    const float* __restrict__ cn2, float* __restrict__ S,
    float* __restrict__ colsum)
{
    __shared__ float lds_logit[16][KK];
    __shared__ float lds_dotc [16][KK];
    __shared__ float lds_s    [16][KK];
    __shared__ float lds_xn   [32];

    const int tile  = blockIdx.x;          // 0..195, 49 tiles per batch
    const int pbase = tile * 16;
    const int batch = tile / 49;
    const int lane  = threadIdx.x & 31;
    const int wave  = threadIdx.x >> 5;    // 0 -> W_assign, 1 -> centers^T

    const int m  = lane & 15;              // A row == B col index within tile
    const int kb = (lane >> 4) * 2;        // K pair base for this lane half

    v8f acc0 = {}, acc1 = {}, acc2 = {}, acc3 = {};
    float xn_part = 0.f;

    const float* xp = x + (size_t)(pbase + m) * DD + kb;

    if (wave == 0) {
        // B = W_assign, element (d,n) at Wa[d*KK + n]; pairs stride KK apart
        const float* bp0 = Wa + (size_t)kb * KK + m;
        gemm16_loop<KK>(xp, bp0, bp0 + 16, bp0 + 32, bp0 + 48,
                        acc0, acc1, acc2, acc3, xn_part);
    } else {
        // B = centers^T, element (d,n) at cen[n*DD + d]; pairs contiguous
        const float* bp0 = cen + (size_t)m * DD + kb;
        gemm16_loop<1>(xp, bp0, bp0 + 16 * DD, bp0 + 32 * DD, bp0 + 48 * DD,
                       acc0, acc1, acc2, acc3, xn_part);
    }

    // scatter accumulators to LDS.  C/D layout: vgpr r -> M = r + 8*(lane>=16)
    {
        float (*dst)[KK] = (wave == 0) ? lds_logit : lds_dotc;
        const int mm_hi = 8 * (lane >> 4);
        const int nn    = lane & 15;
#pragma unroll
        for (int r = 0; r < 8; ++r) {
            const int mm = r + mm_hi;
            dst[mm][nn]      = acc0[r];
            dst[mm][16 + nn] = acc1[r];
            dst[mm][32 + nn] = acc2[r];
            dst[mm][48 + nn] = acc3[r];
        }
        if (wave == 0) lds_xn[lane] = xn_part;
    }
    __syncthreads();

    // phase 2: 64 threads, 4 per row, 16 clusters each
    const int row = threadIdx.x >> 2;
    const int seg = threadIdx.x & 3;
    const float xn = lds_xn[row] + lds_xn[row + 16];

    float lg[16];
    float mx = -INFINITY;
#pragma unroll
    for (int j = 0; j < 16; ++j) {
        const int k = seg * 16 + j;
        const float l = lds_logit[row][k] + ba[k];
        lg[j] = l;
        mx = fmaxf(mx, l);
    }
    mx = fmaxf(mx, __shfl_xor(mx, 1, 32));
    mx = fmaxf(mx, __shfl_xor(mx, 2, 32));

    float ex[16];
    float sum = 0.f;
#pragma unroll
    for (int j = 0; j < 16; ++j) {
        ex[j] = __expf(lg[j] - mx);
        sum += ex[j];
    }
    sum += __shfl_xor(sum, 1, 32);
    sum += __shfl_xor(sum, 2, 32);
    const float inv = 1.f / sum;

#pragma unroll
    for (int j = 0; j < 16; ++j) {
        const int k = seg * 16 + j;
        const float a  = ex[j] * inv;                           // softmax weight
        float n2 = xn - 2.f * lds_dotc[row][k] + cn2[k];        // ||x - c_k||^2
        n2 = fmaxf(n2, 0.f);
        const float s = a * rsqrtf(fmaxf(a * a * n2, EPSV));    // residual scale
        S[(size_t)(pbase + row) * KK + k] = s;
        lds_s[row][k] = s;
    }
    __syncthreads();

    // per-(batch,k) column sums of S tile
    {
        const int k = threadIdx.x;   // 0..63
        float cs = 0.f;
#pragma unroll
        for (int r = 0; r < 16; ++r) cs += lds_s[r][k];
        atomicAdd(&colsum[batch * KK + k], cs);
    }
}

// ---------------------------------------------------------------------------
// Kernel 2: per batch, v = S^T * X - colsum .* centers   (64 x 512), WMMA,
// plus per-batch sum-of-squares for final L2 normalization.
// grid = (d-tiles=32, m-tiles=4, batch=4), one wave per block.
// Software-pipelined over the 784-pixel K dimension.
// ---------------------------------------------------------------------------
__global__ __launch_bounds__(32) void vlad_aggregate_kernel(
    const float* __restrict__ x,      const float* __restrict__ cen,
    const float* __restrict__ S,      const float* __restrict__ colsum,
    float* __restrict__ v,            float* __restrict__ ssq)
{
    const int dtile = blockIdx.x;     // 0..31
    const int mtile = blockIdx.y;     // 0..3
    const int b     = blockIdx.z;     // 0..3
    const int lane  = threadIdx.x;

    const int mn = lane & 15;          // A row (cluster) == B col (d) index
    const int kb = (lane >> 4) * 2;    // pixel pair base
    const int d0 = dtile * 16;

    // per-lane base pointers; iterate p0 in steps of 4 pixels
    const float* Sp = S + (size_t)b * HW * KK + (size_t)kb * KK + mtile * 16 + mn;
    const float* Xp = x + (size_t)b * HW * DD + (size_t)kb * DD + d0 + mn;

    v8f acc = {};
    v2f a, bb;
    a.x  = Sp[0];   a.y  = Sp[KK];
    bb.x = Xp[0];   bb.y = Xp[DD];

    for (int p0 = 0; p0 < HW - 4; p0 += 4) {
        v2f an, bn;
        const size_t pn = (size_t)(p0 + 4);
        an.x = Sp[pn * KK];       an.y = Sp[pn * KK + KK];
        bn.x = Xp[pn * DD];       bn.y = Xp[pn * DD + DD];

        acc = __builtin_amdgcn_wmma_f32_16x16x4_f32(false, a, false, bb, (short)0, acc, false, false);
        a = an; bb = bn;
    }
    acc = __builtin_amdgcn_wmma_f32_16x16x4_f32(false, a, false, bb, (short)0, acc, false, false);

    float sq = 0.f;
    const int mhi = 8 * (lane >> 4);
    const int ng  = d0 + (lane & 15);
#pragma unroll
    for (int r = 0; r < 8; ++r) {
        const int mg = mtile * 16 + r + mhi;                       // cluster id
        const float val = acc[r] - colsum[b * KK + mg] * cen[(size_t)mg * DD + ng];
        v[((size_t)b * KK + mg) * DD + ng] = val;
        sq += val * val;
    }
#pragma unroll
    for (int off = 16; off > 0; off >>= 1) sq += __shfl_xor(sq, off, 32);
    if (lane == 0) atomicAdd(&ssq[b], sq);
}

// ---------------------------------------------------------------------------
// Kernel 3: out = v * rsqrt(max(ssq_b, EPS))
// ---------------------------------------------------------------------------
__global__ __launch_bounds__(256) void vlad_norm_kernel(
    const float* __restrict__ v, const float* __restrict__ ssq,
    float* __restrict__ out)
{
    const int i = blockIdx.x * blockDim.x + threadIdx.x;   // 0 .. 131071
    const int b = i >> 15;                                  // 32768 per batch
    out[i] = v[i] * rsqrtf(fmaxf(ssq[b], EPSV));
}

// ---------------------------------------------------------------------------
extern "C" void kernel_launch(void* const* d_in, const int* in_sizes, int n_in,
                              void* d_out, int out_size, void* d_ws, size_t ws_size,
                              hipStream_t stream) {
    const float* x   = (const float*)d_in[0];   // (4,28,28,512)
    const float* Wa  = (const float*)d_in[1];   // (512,64)
    const float* ba  = (const float*)d_in[2];   // (64,)
    const float* cen = (const float*)d_in[3];   // (64,512)
    float* out = (float*)d_out;                 // (4, 32768)

    // workspace layout (floats)
    float* ws     = (float*)d_ws;
    float* S      = ws;                         // 3136*64       = 200704
    float* cn2    = ws + 200704;                // 64
    float* colsum = ws + 200768;                // 256
    float* ssq    = ws + 201024;                // 4 (+4 pad)
    float* v      = ws + 201032;                // 4*64*512      = 131072

    vlad_prep_kernel<<<1, 256, 0, stream>>>(cen, cn2, colsum, ssq);
    vlad_assign_kernel<<<NPIX / 16, 64, 0, stream>>>(x, Wa, ba, cen, cn2, S, colsum);
    vlad_aggregate_kernel<<<dim3(DD / 16, KK / 16, BATCH), 32, 0, stream>>>(
        x, cen, S, colsum, v, ssq);
    vlad_norm_kernel<<<(BATCH * KK * DD) / 256, 256, 0, stream>>>(v, ssq, out);
}